// GraphDiffusionConvLayer_10385230921953
// MI455X (gfx1250) — compile-verified
//
#include <hip/hip_runtime.h>
#include <hip/hip_bf16.h>

typedef float v2f __attribute__((ext_vector_type(2)));
typedef float v8f __attribute__((ext_vector_type(8)));

#define N_NODES 8192
#define DIM 512
#define WORDS_PER_ROW 256   // 8192 bits / 32

// ---------------------------------------------------------------------------
// Zero the adjacency bitmask (must run every call: atomicOr accumulates).
__global__ void k_zero_bits(unsigned int* __restrict__ bits) {
    bits[(size_t)blockIdx.x * blockDim.x + threadIdx.x] = 0u;
}

// ---------------------------------------------------------------------------
// Scatter edges into symmetric bitmask with SET semantics (dedup via OR).
__global__ void k_edges(const int* __restrict__ ei, int E,
                        unsigned int* __restrict__ bits) {
    int e = blockIdx.x * blockDim.x + threadIdx.x;
    if (e >= E) return;
    int r = ei[e];
    int c = ei[E + e];
    atomicOr(&bits[(size_t)r * WORDS_PER_ROW + (c >> 5)], 1u << (c & 31));
    atomicOr(&bits[(size_t)c * WORDS_PER_ROW + (r >> 5)], 1u << (r & 31));
}

// ---------------------------------------------------------------------------
// inv_norm[i] = 1 / (popcount(row_i) + 1)   (the +1 is the identity's diag;
// max(row_sum,1) in the reference is vacuous since row_sum >= 1 always).
__global__ void k_deg(const unsigned int* __restrict__ bits,
                      float* __restrict__ inv_norm) {
    int i = blockIdx.x * blockDim.x + threadIdx.x;
    if (i >= N_NODES) return;
    const unsigned int* row = bits + (size_t)i * WORDS_PER_ROW;
    int cnt = 1;
    for (int w = 0; w < WORDS_PER_ROW; ++w) cnt += __popc(row[w]);
    inv_norm[i] = 1.0f / (float)cnt;
}

// ---------------------------------------------------------------------------
// h = x @ W + b  via full-precision V_WMMA_F32_16X16X4_F32.
// One wave per 16x16 output tile; 128 WMMAs along K=512.
// Epilogue also seeds the diffusion accumulator: acc = dw[0] * h.
__global__ void k_gemm_wmma(const float* __restrict__ x,
                            const float* __restrict__ W,
                            const float* __restrict__ bias,
                            const float* __restrict__ dw,
                            float* __restrict__ h,
                            float* __restrict__ acc) {
    const int lane = threadIdx.x & 31;
    const int wave = threadIdx.x >> 5;
    const int tile = blockIdx.x * 8 + wave;      // 16384 tiles total
    const int tm = tile >> 5;                    // 512 row tiles
    const int tn = tile & 31;                    // 32  col tiles
    const int l16 = lane & 15;
    const int lh  = lane >> 4;                   // half-wave selects K pair

    const float* xr = x + ((size_t)(tm * 16 + l16)) * DIM;   // A row for this lane
    const float* wc = W + tn * 16 + l16;                     // B column for this lane

    v8f c = {0.f, 0.f, 0.f, 0.f, 0.f, 0.f, 0.f, 0.f};
    for (int kk = 0; kk < DIM; kk += 4) {
        const int ka = kk + lh * 2;
        v2f a, b;
        a.x = xr[ka];
        a.y = xr[ka + 1];
        b.x = wc[(size_t)ka * DIM];
        b.y = wc[(size_t)(ka + 1) * DIM];
        c = __builtin_amdgcn_wmma_f32_16x16x4_f32(
                /*neg_a=*/false, a, /*neg_b=*/false, b,
                /*c_mod=*/(short)0, c, /*reuse_a=*/false, /*reuse_b=*/false);
    }

    const float w0 = dw[0];
    const int col = tn * 16 + l16;
    const float bc = bias[col];
#pragma unroll
    for (int r = 0; r < 8; ++r) {
        const int row = tm * 16 + lh * 8 + r;    // C layout: lanes 16-31 hold M+8
        const float v = c[r] + bc;
        const size_t idx = (size_t)row * DIM + col;
        h[idx]   = v;
        acc[idx] = w0 * v;
    }
}

// ---------------------------------------------------------------------------
// One block per node row i:
//   nxt[i,:] = inv_norm[i] * (cur[i,:] + sum_{j: bit(i,j)} cur[j,:])
//   acc[i,:] += dw[widx] * nxt[i,:]     (+ ReLU on the final step)
// 256 threads, each owns 2 columns (float2) -> coalesced 2KB gathers per
// neighbor row, served from L2 (cur buffer is L2-resident).
__global__ void k_spmm(const unsigned int* __restrict__ bits,
                       const float* __restrict__ inv_norm,
                       const float* __restrict__ cur,
                       const float* __restrict__ dw, int widx, int last,
                       float* __restrict__ nxt,
                       float* __restrict__ acc) {
    __shared__ unsigned int srow[WORDS_PER_ROW];
    const int i = blockIdx.x;
    const int t = threadIdx.x;
    srow[t] = bits[(size_t)i * WORDS_PER_ROW + t];
    __syncthreads();

    const int col = t * 2;
    const size_t self = (size_t)i * DIM + col;
    float s0 = cur[self];          // identity (self-loop from +eye)
    float s1 = cur[self + 1];

    for (int w = 0; w < WORDS_PER_ROW; ++w) {
        unsigned int m = srow[w];
        while (m) {
            const int b = __builtin_ctz(m);
            m &= m - 1;
            const size_t jbase = (size_t)(w * 32 + b) * DIM + col;
            s0 += cur[jbase];
            s1 += cur[jbase + 1];
        }
    }

    const float sc = inv_norm[i];
    s0 *= sc; s1 *= sc;
    nxt[self]     = s0;
    nxt[self + 1] = s1;

    const float wgt = dw[widx];
    float o0 = acc[self]     + wgt * s0;
    float o1 = acc[self + 1] + wgt * s1;
    if (last) { o0 = fmaxf(o0, 0.0f); o1 = fmaxf(o1, 0.0f); }
    acc[self]     = o0;
    acc[self + 1] = o1;
}

// ---------------------------------------------------------------------------
extern "C" void kernel_launch(void* const* d_in, const int* in_sizes, int n_in,
                              void* d_out, int out_size, void* d_ws, size_t ws_size,
                              hipStream_t stream) {
    const float* x    = (const float*)d_in[0];
    const int*   ei   = (const int*)d_in[1];
    const float* W    = (const float*)d_in[2];
    const float* bias = (const float*)d_in[3];
    const float* dw   = (const float*)d_in[4];
    float* out = (float*)d_out;

    const int E = in_sizes[1] / 2;

    // Workspace layout: bits (8MB) | inv_norm (32KB) | bufA (16MB) | bufB (16MB)
    unsigned int* bits = (unsigned int*)d_ws;
    float* inv  = (float*)(bits + (size_t)N_NODES * WORDS_PER_ROW);
    float* bufA = inv + N_NODES;
    float* bufB = bufA + (size_t)N_NODES * DIM;

    // 1. Rebuild adjacency bitmask from scratch each call (determinism).
    k_zero_bits<<<(N_NODES * WORDS_PER_ROW) / 256, 256, 0, stream>>>(bits);
    k_edges<<<(E + 255) / 256, 256, 0, stream>>>(ei, E, bits);
    k_deg<<<N_NODES / 256, 256, 0, stream>>>(bits, inv);

    // 2. h = xW + b (FP32 WMMA); acc(d_out) = dw[0]*h.
    k_gemm_wmma<<<(N_NODES / 16) * (DIM / 16) / 8, 256, 0, stream>>>(
        x, W, bias, dw, bufA, out);

    // 3. Three diffusion hops, ping-ponging bufA/bufB; ReLU fused in last.
    k_spmm<<<N_NODES, 256, 0, stream>>>(bits, inv, bufA, dw, 1, 0, bufB, out);
    k_spmm<<<N_NODES, 256, 0, stream>>>(bits, inv, bufB, dw, 2, 0, bufA, out);
    k_spmm<<<N_NODES, 256, 0, stream>>>(bits, inv, bufA, dw, 3, 1, bufB, out);
}